// Gemma4VisionPooler_86320252715806
// MI455X (gfx1250) — compile-verified
//
#include <hip/hip_runtime.h>
#include <hip/hip_bf16.h>
#include <stdint.h>

// Problem constants (reference: B=16, S=4096, D=1152, L=1024, k=2)
#define BB 16
#define SS 4096
#define DD 1152
#define LL 1024
#define KK 2
// ROOT / k^2 = sqrt(1152) / 4
#define SCALE_F (33.941125496954285f * 0.25f)

typedef float v4f __attribute__((ext_vector_type(4)));

// ---------------------------------------------------------------------------
// 1) zero the small integer scratch (bucket counts + per-batch max-x)
// ---------------------------------------------------------------------------
__global__ void k_init_small(int* __restrict__ cnt, int* __restrict__ maxx) {
    int i = blockIdx.x * blockDim.x + threadIdx.x;
    if (i < BB * LL) cnt[i] = 0;
    if (i < BB)      maxx[i] = 0;
}

// ---------------------------------------------------------------------------
// 2) zero pooled output (we accumulate with atomics; harness does not re-zero)
// ---------------------------------------------------------------------------
__global__ void k_zero_out(v4f* __restrict__ out4, int n4) {
    int i = blockIdx.x * blockDim.x + threadIdx.x;
    int stride = gridDim.x * blockDim.x;
    v4f z = {0.f, 0.f, 0.f, 0.f};
    for (; i < n4; i += stride) out4[i] = z;
}

// ---------------------------------------------------------------------------
// 3) per-batch max over clamped x positions (wave32 reduce, then atomicMax)
//    token t in [0, B*S); batch = t >> 12 since S = 4096
// ---------------------------------------------------------------------------
__global__ void k_maxx(const int* __restrict__ pos, int* __restrict__ maxx) {
    int t = blockIdx.x * blockDim.x + threadIdx.x;
    int b = t >> 12;
    int x = pos[t * 2];
    x = x < 0 ? 0 : x;
#pragma unroll
    for (int o = 16; o > 0; o >>= 1) {
        int y = __shfl_xor(x, o, 32);
        x = x > y ? x : y;
    }
    if ((threadIdx.x & 31) == 0) atomicMax(&maxx[b], x);
}

// ---------------------------------------------------------------------------
// 4) bucket index per token + per-bucket population count
// ---------------------------------------------------------------------------
__global__ void k_idx_cnt(const int* __restrict__ pos, const int* __restrict__ maxx,
                          int* __restrict__ idx, int* __restrict__ cnt) {
    int t = blockIdx.x * blockDim.x + threadIdx.x;
    int b = t >> 12;
    int x = pos[t * 2 + 0];
    int y = pos[t * 2 + 1];
    x = x < 0 ? 0 : x;
    y = y < 0 ? 0 : y;
    int stride = (maxx[b] + 1) / KK;           // (max_x // k)
    int id = (x / KK) + stride * (y / KK);
    if (id > LL - 1) id = LL - 1;              // safety clamp
    idx[t] = id;
    atomicAdd(&cnt[b * LL + id], 1);
}

// ---------------------------------------------------------------------------
// 5) scatter-add, double-buffered through LDS with CDNA5 async copies.
//    Block = 288 threads (9 wave32s); lane i owns 16B of the 4608B token row.
//    Input is streamed with th:TH_LOAD_NT (302 MB > 192 MB L2) so the
//    L2-resident 75 MB output stays hot for the global_atomic_add_f32 ops.
//    Pipeline: async-load row(t+1) into other stage, s_wait_asynccnt<=1,
//    consume row(t) from LDS, 4x global_atomic_add_f32.
// ---------------------------------------------------------------------------
__global__ __launch_bounds__(288)
void k_scatter(const float* __restrict__ h, const int* __restrict__ idx,
               const unsigned char* __restrict__ pad, float* __restrict__ out) {
    __shared__ v4f buf[2][288];

    const int      tid   = threadIdx.x;
    const uint64_t hbase = (uint64_t)(uintptr_t)h;                 // SGPR pair (GVS saddr)
    const uint32_t lds0  = (uint32_t)(uintptr_t)&buf[0][tid];      // per-lane LDS byte addr
    const uint32_t lds1  = (uint32_t)(uintptr_t)&buf[1][tid];
    const int      total = BB * SS;

    int t = blockIdx.x;
    if (t < total) {
        uint32_t goff = (uint32_t)t * (DD * 4u) + (uint32_t)tid * 16u;
        asm volatile("global_load_async_to_lds_b128 %0, %1, %2 th:TH_LOAD_NT"
                     :: "v"(lds0), "v"(goff), "s"(hbase) : "memory");
    }

    int stage = 0;
    for (; t < total; t += (int)gridDim.x) {
        int tn = t + (int)gridDim.x;
        if (tn < total) {
            uint32_t goff = (uint32_t)tn * (DD * 4u) + (uint32_t)tid * 16u;
            uint32_t lds  = stage ? lds0 : lds1;
            asm volatile("global_load_async_to_lds_b128 %0, %1, %2 th:TH_LOAD_NT"
                         :: "v"(lds), "v"(goff), "s"(hbase) : "memory");
            asm volatile("s_wait_asynccnt 0x1" ::: "memory");      // row t landed
        } else {
            asm volatile("s_wait_asynccnt 0x0" ::: "memory");
        }

        int b  = t >> 12;
        int id = idx[t];
        if (!pad[t]) {
            // volatile: LDS was written by the async engine, not by this wave's stores
            v4f v = *(volatile const v4f*)&buf[stage][tid];
            float* o = out + (size_t)(b * LL + id) * DD + tid * 4;
            unsafeAtomicAdd(o + 0, v.x * SCALE_F);
            unsafeAtomicAdd(o + 1, v.y * SCALE_F);
            unsafeAtomicAdd(o + 2, v.z * SCALE_F);
            unsafeAtomicAdd(o + 3, v.w * SCALE_F);
        }
        stage ^= 1;
    }
}

// ---------------------------------------------------------------------------
// 6) nonempty mask (stored as 0.0/1.0 floats after the pooled block)
// ---------------------------------------------------------------------------
__global__ void k_mask(const int* __restrict__ cnt, float* __restrict__ m) {
    int i = blockIdx.x * blockDim.x + threadIdx.x;
    if (i < BB * LL) m[i] = cnt[i] > 0 ? 1.0f : 0.0f;
}

// ---------------------------------------------------------------------------
// launcher
//   d_in[0] hidden_states      float32  [B,S,D]
//   d_in[1] pixel_position_ids int32    [B,S,2]
//   d_in[2] padding_positions  bool     [B,S]
//   d_in[3] output_length      (scalar, hardcoded 1024)
//   d_out:  pooled float32 [B,L,D] then mask [B,L]
//   d_ws:   cnt[B*L] ints | maxx[B] ints | idx[B*S] ints  (~320 KB)
// ---------------------------------------------------------------------------
extern "C" void kernel_launch(void* const* d_in, const int* in_sizes, int n_in,
                              void* d_out, int out_size, void* d_ws, size_t ws_size,
                              hipStream_t stream) {
    (void)in_sizes; (void)n_in; (void)out_size; (void)ws_size;

    const float*         h   = (const float*)d_in[0];
    const int*           pos = (const int*)d_in[1];
    const unsigned char* pad = (const unsigned char*)d_in[2];
    float*               out = (float*)d_out;

    int* ws_cnt  = (int*)d_ws;           // B*L
    int* ws_maxx = ws_cnt + BB * LL;     // B
    int* ws_idx  = ws_maxx + BB;         // B*S

    k_init_small<<<(BB * LL + 255) / 256, 256, 0, stream>>>(ws_cnt, ws_maxx);
    k_zero_out<<<4096, 256, 0, stream>>>((v4f*)out, (BB * LL * DD) / 4);
    k_maxx<<<(BB * SS) / 256, 256, 0, stream>>>(pos, ws_maxx);
    k_idx_cnt<<<(BB * SS) / 256, 256, 0, stream>>>(pos, ws_maxx, ws_idx, ws_cnt);
    k_scatter<<<4096, 288, 0, stream>>>(h, ws_idx, pad, out);
    k_mask<<<(BB * LL + 255) / 256, 256, 0, stream>>>(ws_cnt, out + (size_t)BB * LL * DD);
}